// BagEmbedding_82222853914904
// MI455X (gfx1250) — compile-verified
//
#include <hip/hip_runtime.h>
#include <stdint.h>

// BagEmbedding gather-reduce for MI455X (gfx1250).
// out[r, :] = sum_w (X[r,w]!=0) * W[X[r,w], :],  r in [0, 8192), 50 words, 128 dims.
// Strategy: TDM gather-mode descriptors (8 rows / 32-bit indices each) DMA the
// embedding rows into a per-wave double-buffered LDS tile; the wave pipelines
// issue/consume with TENSORcnt. Indices are wave-uniform -> scalar SMEM loads
// (compiler fuses them into s_load_b512). Consume: 8 batched ds_load_b128 per
// chunk, masked accumulate, one global_store_b128 per lane at the end.

#define VOCAB      100000
#define EMBED      128
#define NB_WORDS   50
#define ROWS       8192           // BATCH(64) * INPUT_LEN(128)
#define WAVES_PB   8              // waves per block (256 threads, wave32)
#define CHUNK      8              // rows per TDM gather descriptor (32-bit idx max)
#define NCHUNK     7              // ceil(50 / 8)
#define ROW_BYTES  (EMBED * 4)    // 512 B
#define BUF_BYTES  (CHUNK * ROW_BYTES) // 4 KB per buffer

typedef __attribute__((ext_vector_type(4))) unsigned int u32x4;
typedef __attribute__((ext_vector_type(8))) int          i32x8;
typedef __attribute__((ext_vector_type(4))) int          i32x4;
typedef __attribute__((ext_vector_type(4))) float        f32x4;

#if __has_builtin(__builtin_amdgcn_tensor_load_to_lds)
#define HAVE_TDM 1
#else
#define HAVE_TDM 0
#endif

#if HAVE_TDM
__global__ __launch_bounds__(WAVES_PB * 32)
void bag_embed_tdm(const int* __restrict__ X, const float* __restrict__ W,
                   float* __restrict__ out) {
  // [wave][buf][row][embed] : 8 * 2 * 8 * 128 * 4B = 64 KB (WGP has 320 KB,
  // so 5 blocks/WGP = 40 waves/WGP alongside the per-wave TDM double buffer).
  __shared__ float lds[WAVES_PB][2][CHUNK][EMBED];

  const int lane = threadIdx.x & 31;
  const int wave = __builtin_amdgcn_readfirstlane((int)(threadIdx.x >> 5));
  const int row  = blockIdx.x * WAVES_PB + wave;         // 0..8191 exactly

  const int* xr = X + row * NB_WORDS;                    // wave-uniform -> s_load

  const unsigned long long wa = (unsigned long long)(uintptr_t)W;
  const unsigned int wa_lo = (unsigned int)wa;
  const unsigned int wa_hi = (unsigned int)(wa >> 32);

  // Flat shared pointer is {aperture_hi32, lds_offset32}; low 32 bits = LDS byte
  // offset within the workgroup allocation (hardware adds wave LDS_BASE).
  const unsigned int lds_base =
      (unsigned int)(uintptr_t)(void*)&lds[wave][0][0][0];

  int idxbuf[2][CHUNK];

  auto issue = [&](int c) {
    const int n    = (c == NCHUNK - 1) ? (NB_WORDS - (NCHUNK - 1) * CHUNK) : CHUNK;
    const int base = c * CHUNK;
    int* ib = idxbuf[c & 1];
#pragma unroll
    for (int j = 0; j < CHUNK; ++j)
      ib[j] = (j < n) ? xr[base + j] : 0;                // pad slots -> index 0

    // ---- D# group 0 (4 dwords) ----
    u32x4 g0;
    g0[0] = 0xC0000001u;                   // count=1 | gather_index_size=32b | gather_mode
    g0[1] = lds_base + (unsigned)(c & 1) * BUF_BYTES;    // lds_addr
    g0[2] = wa_lo;                                       // global_addr[31:0]
    g0[3] = (wa_hi & 0x01FFFFFFu) | 0x80000000u;         // addr[56:32] | type=2

    // ---- D# group 1 (8 dwords) ----
    i32x8 g1;
    g1[0] = 0x00020000;                    // workgroup_mask=0, data_size=4B
    g1[1] = (int)((unsigned)EMBED << 16);  // tensor_dim0=128 (low16 in [31:16])
    g1[2] = (int)0x86A00000u;              // tensor_dim1=100000 low16 in [31:16]
    g1[3] = (int)(((unsigned)EMBED << 16) | 1u); // tile_dim0=128 | tensor_dim1 hi16
    g1[4] = n;                             // tile_dim1 = #valid gather indices
    g1[5] = EMBED;                         // tensor_dim0_stride = 128 elements
    g1[6] = 0;                             // stride hi / dim1_stride (ignored: gather)
    g1[7] = 0;

    // ---- D# groups 2/3: 32-bit row indices (up to 8) ----
    i32x4 g2 = { ib[0], ib[1], ib[2], ib[3] };
    i32x4 g3 = { ib[4], ib[5], ib[6], ib[7] };

#if defined(__clang_major__) && __clang_major__ >= 23
    i32x8 gx = {};                         // clang-23 therock headers: 6-arg form
    __builtin_amdgcn_tensor_load_to_lds(g0, g1, g2, g3, gx, 0);
#else
    __builtin_amdgcn_tensor_load_to_lds(g0, g1, g2, g3, 0);   // ROCm 7.2: 5-arg
#endif
  };

  f32x4 acc = {0.f, 0.f, 0.f, 0.f};

  issue(0);
#pragma unroll
  for (int c = 0; c < NCHUNK; ++c) {
    if (c + 1 < NCHUNK) {
      issue(c + 1);
      __builtin_amdgcn_s_wait_tensorcnt(1);   // in-order per wave: chunk c complete
    } else {
      __builtin_amdgcn_s_wait_tensorcnt(0);
    }

    const int* ib = idxbuf[c & 1];
    const float* buf = &lds[wave][c & 1][0][lane * 4];

    // Batch all 8 ds_load_b128 (unconditional) so they issue back-to-back and
    // drain in-order; then mask the accumulate with a wave-uniform branch.
    // Pad rows always have ib[j]==0, so unwritten LDS never feeds arithmetic.
    f32x4 v[CHUNK];
#pragma unroll
    for (int j = 0; j < CHUNK; ++j)
      v[j] = *(const f32x4*)(buf + j * EMBED);
#pragma unroll
    for (int j = 0; j < CHUNK; ++j)
      if (ib[j] != 0) acc += v[j];
  }

  *(f32x4*)(out + row * EMBED + lane * 4) = acc;          // global_store_b128
}
#else
// Fallback: direct vectorized gather (global_load_b128 per lane, 512B/row/wave).
__global__ __launch_bounds__(WAVES_PB * 32)
void bag_embed_direct(const int* __restrict__ X, const float* __restrict__ W,
                      float* __restrict__ out) {
  const int lane = threadIdx.x & 31;
  const int wave = __builtin_amdgcn_readfirstlane((int)(threadIdx.x >> 5));
  const int row  = blockIdx.x * WAVES_PB + wave;
  const int* xr  = X + row * NB_WORDS;

  f32x4 acc = {0.f, 0.f, 0.f, 0.f};
#pragma unroll 5
  for (int w = 0; w < NB_WORDS; ++w) {
    const int idx = xr[w];                                // wave-uniform s_load
    if (idx != 0) {
      const f32x4 v = *(const f32x4*)(W + (size_t)idx * EMBED + lane * 4);
      acc += v;
    }
  }
  *(f32x4*)(out + row * EMBED + lane * 4) = acc;
}
#endif

extern "C" void kernel_launch(void* const* d_in, const int* in_sizes, int n_in,
                              void* d_out, int out_size, void* d_ws, size_t ws_size,
                              hipStream_t stream) {
  const int*   X   = (const int*)d_in[0];    // (64,128,50) int32
  const float* W   = (const float*)d_in[1];  // (100000,128) float32
  float*       out = (float*)d_out;          // (64,128,128) float32

  dim3 grid(ROWS / WAVES_PB);                // 1024 blocks
  dim3 block(WAVES_PB * 32);                 // 256 threads = 8 wave32
#if HAVE_TDM
  bag_embed_tdm<<<grid, block, 0, stream>>>(X, W, out);
#else
  bag_embed_direct<<<grid, block, 0, stream>>>(X, W, out);
#endif
}